// Constrained_Affine_5188320493997
// MI455X (gfx1250) — compile-verified
//
#include <hip/hip_runtime.h>
#include <hip/hip_bf16.h>
#include <math.h>

#define N_IMGS 32
#define N_ROT  17          // S_ + 1
#define N_CH   3
#define IMG_H  224
#define IMG_W  224
#define TILE   32
#define TPB    256
#define STG    48          // staged input window side (covers 31*sqrt(2)+3)
#define STG_STRIDE 49      // padded LDS row stride in floats
#define STG_ELEMS (STG * STG)                 // 2304 texels per channel
#define LDS_FLOATS (N_CH * STG * STG_STRIDE)  // 7056 floats = 28224 B

// gfx1250 async global->LDS path (guarded so host pass / older toolchains still compile)
#if defined(__AMDGCN__) && __has_builtin(__builtin_amdgcn_global_load_async_to_lds_b32) && \
    __has_builtin(__builtin_amdgcn_s_wait_asynccnt)
#define USE_ASYNC_LDS 1
#else
#define USE_ASYNC_LDS 0
#endif

extern "C" __global__ __launch_bounds__(TPB) void rot_bilinear_sample_kernel(
    const float* __restrict__ x,       // [N][C][H][W]
    const float* __restrict__ thetas,  // [N][N_ROT]
    float* __restrict__ out)           // [N][N_ROT][C][H][W]
{
    extern __shared__ float smem[];    // [N_CH][STG][STG_STRIDE], dynamic => LDS offset 0

    const int tile = blockIdx.x;                 // 0..48
    const int s    = blockIdx.y;                 // 0..16
    const int n    = blockIdx.z;                 // 0..31
    const int ty   = tile / (IMG_W / TILE);
    const int tx   = tile - ty * (IMG_W / TILE);
    const int h0   = ty * TILE;
    const int w0   = tx * TILE;
    const int tid  = threadIdx.x;

    const float th = thetas[n * N_ROT + s];
    const float ct = cosf(th);
    const float st = sinf(th);

    // output pixel (w,h) -> input pixel coords:
    //   xs=(w+0.5)*2/W-1 ; ys=(h+0.5)*2/H-1
    //   gx=ct*xs-st*ys ; gy=st*xs+ct*ys
    //   ix=((gx+1)*W-1)/2 ; iy=((gy+1)*H-1)/2
    // Affine in (w,h) -> footprint extrema at the 4 tile corners.
    float min_ix = 1e30f, min_iy = 1e30f;
#pragma unroll
    for (int cy = 0; cy < 2; ++cy) {
#pragma unroll
        for (int cx = 0; cx < 2; ++cx) {
            float wf = (float)(w0 + cx * (TILE - 1));
            float hf = (float)(h0 + cy * (TILE - 1));
            float xs = (wf + 0.5f) * (2.0f / IMG_W) - 1.0f;
            float ys = (hf + 0.5f) * (2.0f / IMG_H) - 1.0f;
            float gx = ct * xs - st * ys;
            float gy = st * xs + ct * ys;
            float ix = ((gx + 1.0f) * IMG_W - 1.0f) * 0.5f;
            float iy = ((gy + 1.0f) * IMG_H - 1.0f) * 0.5f;
            min_ix = fminf(min_ix, ix);
            min_iy = fminf(min_iy, iy);
        }
    }
    const int x0 = (int)floorf(min_ix) - 1;   // 1-texel safety margin; window width 48 covers
    const int y0 = (int)floorf(min_iy) - 1;   // floor-span (<=45) + margin for any rotation

    // ---- Stage 48x48x3 input window into LDS (zeros outside the image => zeros padding) ----
    const float* img = x + (size_t)n * N_CH * IMG_H * IMG_W;
    for (int e = tid; e < STG_ELEMS; e += TPB) {     // 2304/256 = 9 iterations
        int r  = e / STG;
        int cl = e - r * STG;
        int yy = y0 + r;
        int xx = x0 + cl;
        bool inimg = (xx >= 0) & (xx < IMG_W) & (yy >= 0) & (yy < IMG_H);
#pragma unroll
        for (int ch = 0; ch < N_CH; ++ch) {
            int li = (ch * STG + r) * STG_STRIDE + cl;
            if (inimg) {
#if USE_ASYNC_LDS
                // AS3 pointer = low 32 bits of the generic LDS address (flat-aperture rule);
                // integer round-trips avoid illegal generic->AS casts.
                unsigned lofs = (unsigned)(unsigned long long)(const void*)&smem[li];
                __attribute__((address_space(3))) int* lp =
                    (__attribute__((address_space(3))) int*)(unsigned long long)lofs;
                __attribute__((address_space(1))) int* gp =
                    (__attribute__((address_space(1))) int*)(unsigned long long)
                        (const void*)(img + ((size_t)ch * IMG_H + yy) * IMG_W + xx);
                __builtin_amdgcn_global_load_async_to_lds_b32(gp, lp, 0, 0);
#else
                smem[li] = img[((size_t)ch * IMG_H + yy) * IMG_W + xx];
#endif
            } else {
                smem[li] = 0.0f;   // zeros padding lives in the staged window
            }
        }
    }
#if USE_ASYNC_LDS
    __builtin_amdgcn_s_wait_asynccnt(0);   // drain this wave's async LDS writes
#endif
    __syncthreads();                        // waits DScnt + workgroup barrier

    // ---- Compute: each thread -> 4 consecutive w pixels x 3 channels, float4 stores ----
    const int wq = (tid & 7) * 4;           // 0,4,...,28 within tile
    const int py = tid >> 3;                // 0..31
    const int h  = h0 + py;
    const float ys = ((float)h + 0.5f) * (2.0f / IMG_H) - 1.0f;

    float acc[N_CH][4];
#pragma unroll
    for (int k = 0; k < 4; ++k) {
        int w = w0 + wq + k;
        float xs = ((float)w + 0.5f) * (2.0f / IMG_W) - 1.0f;
        float gx = ct * xs - st * ys;
        float gy = st * xs + ct * ys;
        float ix = ((gx + 1.0f) * IMG_W - 1.0f) * 0.5f;
        float iy = ((gy + 1.0f) * IMG_H - 1.0f) * 0.5f;
        float fx = floorf(ix);
        float fy = floorf(iy);
        float wx1 = ix - fx, wy1 = iy - fy;
        float wx0 = 1.0f - wx1, wy0 = 1.0f - wy1;
        int lx = (int)fx - x0;              // in [1, 46] by bbox construction
        int ly = (int)fy - y0;
#pragma unroll
        for (int ch = 0; ch < N_CH; ++ch) {
            const float* sp = &smem[(ch * STG + ly) * STG_STRIDE + lx];
            float v00 = sp[0];
            float v01 = sp[1];
            float v10 = sp[STG_STRIDE];
            float v11 = sp[STG_STRIDE + 1];
            acc[ch][k] = wy0 * (wx0 * v00 + wx1 * v01) +
                         wy1 * (wx0 * v10 + wx1 * v11);
        }
    }

    const size_t obase =
        ((((size_t)n * N_ROT + s) * N_CH) * IMG_H + (size_t)h) * IMG_W + (w0 + wq);
#pragma unroll
    for (int ch = 0; ch < N_CH; ++ch) {
        float4 v = make_float4(acc[ch][0], acc[ch][1], acc[ch][2], acc[ch][3]);
        *(float4*)(out + obase + (size_t)ch * IMG_H * IMG_W) = v;   // global_store_b128
    }
}

extern "C" void kernel_launch(void* const* d_in, const int* in_sizes, int n_in,
                              void* d_out, int out_size, void* d_ws, size_t ws_size,
                              hipStream_t stream) {
    (void)in_sizes; (void)n_in; (void)out_size; (void)d_ws; (void)ws_size;
    const float* x      = (const float*)d_in[0];   // [32][3][224][224]
    const float* thetas = (const float*)d_in[1];   // [32][17]
    float* out          = (float*)d_out;           // [32][17][3][224][224]

    dim3 grid((IMG_H / TILE) * (IMG_W / TILE), N_ROT, N_IMGS);   // (49, 17, 32)
    dim3 block(TPB);
    size_t shmem = sizeof(float) * LDS_FLOATS;                   // 28224 bytes
    rot_bilinear_sample_kernel<<<grid, block, shmem, stream>>>(x, thetas, out);
}